// GAT_26036091748786
// MI455X (gfx1250) — compile-verified
//
#include <hip/hip_runtime.h>
#include <math.h>

// ---------------- problem constants (match reference) ----------------
#define NN    30000
#define EE    300000
#define ETOT  330000      // E + N self loops
#define BB    300
#define EMB   64
#define D1    256         // HEADS*EMB
#define OUT3  121
#define H3    6
#define D3    726         // H3*OUT3
#define D3P   736         // padded to multiple of 16
#define HP    8           // per-node/edge head stride (padded)
#define NCLS  10
#define NT    2           // N-tiles per wave in the WMMA GEMM

typedef float v2f __attribute__((ext_vector_type(2)));
typedef float v8f __attribute__((ext_vector_type(8)));

// ---------------- helpers ----------------
__device__ __forceinline__ void edge_sd(const int* __restrict__ ei, int e, int& s, int& d) {
    if (e < EE) { s = ei[e]; d = ei[EE + e]; }
    else        { s = e - EE; d = s; }
}

// ---------------- kernels ----------------
__global__ void fill_k(float* p, float v, long n) {
    long i = (long)blockIdx.x * blockDim.x + threadIdx.x;
    if (i < n) p[i] = v;
}

__global__ void padw3_k(const float* __restrict__ w3, float* __restrict__ w3p) {
    int t = blockIdx.x * blockDim.x + threadIdx.x;
    if (t >= 256 * D3P) return;
    int k = t / D3P, n = t % D3P;
    w3p[t] = (n < D3) ? w3[k * D3 + n] : 0.0f;
}

__global__ void encode_k(const int* __restrict__ x, const int* __restrict__ depth,
                         const float* __restrict__ nemb, const float* __restrict__ demb,
                         float* __restrict__ A) {
    int t = blockIdx.x * blockDim.x + threadIdx.x;
    if (t >= NN * EMB) return;
    int n = t >> 6, d = t & 63;
    A[t] = nemb[x[n] * EMB + d] + demb[depth[n] * EMB + d];
}

// One wave per 16x(NT*16) output strip; fp32 WMMA, K stepped by 4.
// A fragment is loaded once per K-step and reused for NT WMMAs.
__global__ void gemm_wmma_f32(const float* __restrict__ A, const float* __restrict__ W,
                              float* __restrict__ C, int K, int Dpad) {
    int lane = threadIdx.x & 31;
    int wid  = blockIdx.x * (blockDim.x >> 5) + (threadIdx.x >> 5);
    int ngroups = (Dpad >> 4) / NT;        // 256->8, 736->23 (both exact)
    int mt = wid / ngroups;
    int ng = wid - mt * ngroups;
    if (mt * 16 >= NN) return;

    int mrow  = mt * 16 + (lane & 15);
    int kh    = (lane >> 4) << 1;          // 0 or 2 : K sub-row pair for this half-wave
    int ncol  = ng * (NT * 16) + (lane & 15);
    const float* arow = A + (long)mrow * K;

    v8f acc0 = {}, acc1 = {};
    #pragma unroll 4
    for (int k = 0; k < K; k += 4) {
        v2f av, b0, b1;
        av.x = arow[k + kh];
        av.y = arow[k + kh + 1];
        const float* wr0 = W + (long)(k + kh)     * Dpad;
        const float* wr1 = W + (long)(k + kh + 1) * Dpad;
        b0.x = wr0[ncol];      b0.y = wr1[ncol];
        b1.x = wr0[ncol + 16]; b1.y = wr1[ncol + 16];
        acc0 = __builtin_amdgcn_wmma_f32_16x16x4_f32(false, av, false, b0,
                                                     (short)0, acc0, false, false);
        acc1 = __builtin_amdgcn_wmma_f32_16x16x4_f32(false, av, false, b1,
                                                     (short)0, acc1, false, false);
    }
    int rbase = mt * 16 + ((lane >> 4) << 3);
    float* crow = C + (long)rbase * Dpad + ncol;
    #pragma unroll
    for (int r = 0; r < 8; ++r) {
        crow[(long)r * Dpad]      = acc0[r];
        crow[(long)r * Dpad + 16] = acc1[r];
    }
}

__global__ void logits_k(const float* __restrict__ Hf, const float* __restrict__ asrc,
                         const float* __restrict__ adst, float* __restrict__ es,
                         float* __restrict__ ed, int Hh, int OD, int Dpad) {
    int t = blockIdx.x * blockDim.x + threadIdx.x;
    if (t >= NN * Hh) return;
    int n = t / Hh, h = t - n * Hh;
    const float* row = Hf + (long)n * Dpad + h * OD;
    float s = 0.0f, d2 = 0.0f;
    for (int j = 0; j < OD; ++j) {
        float v = row[j];
        s  += v * asrc[h * OD + j];
        d2 += v * adst[h * OD + j];
    }
    es[n * HP + h] = s;
    ed[n * HP + h] = d2;
}

__global__ void edge_max_k(const int* __restrict__ ei, const float* __restrict__ es,
                           const float* __restrict__ ed, float* __restrict__ ebuf,
                           float* __restrict__ mx, int Hh) {
    int t = blockIdx.x * blockDim.x + threadIdx.x;
    if (t >= ETOT * Hh) return;
    int e = t / Hh, h = t - e * Hh;
    int s, d; edge_sd(ei, e, s, d);
    float x = es[s * HP + h] + ed[d * HP + h];
    x = (x > 0.0f) ? x : 0.2f * x;                 // leaky_relu(., 0.2)
    ebuf[(long)e * HP + h] = x;
    // native GLOBAL_ATOMIC_MAX_NUM_F32
    __hip_atomic_fetch_max(&mx[d * HP + h], x, __ATOMIC_RELAXED, __HIP_MEMORY_SCOPE_AGENT);
}

__global__ void edge_exp_k(const int* __restrict__ ei, const float* __restrict__ mx,
                           float* __restrict__ ebuf, float* __restrict__ sx, int Hh) {
    int t = blockIdx.x * blockDim.x + threadIdx.x;
    if (t >= ETOT * Hh) return;
    int e = t / Hh, h = t - e * Hh;
    int s, d; edge_sd(ei, e, s, d);
    float ex = expf(ebuf[(long)e * HP + h] - mx[d * HP + h]);
    ebuf[(long)e * HP + h] = ex;
    atomicAdd(&sx[d * HP + h], ex);
}

// grid.x = edge, grid.y = feature chunk
__global__ void scatter_k(const int* __restrict__ ei, const float* __restrict__ Hf,
                          const float* __restrict__ ebuf, const float* __restrict__ sx,
                          float* __restrict__ O, int OD, int Hh, int Dpad) {
    int e = blockIdx.x;
    int d = blockIdx.y * blockDim.x + threadIdx.x;
    if (d >= OD * Hh) return;
    int s, dn; edge_sd(ei, e, s, dn);
    int h = d / OD;
    float alpha = ebuf[(long)e * HP + h] / (sx[dn * HP + h] + 1e-16f);
    atomicAdd(&O[(long)dn * Dpad + d], alpha * Hf[(long)s * Dpad + d]);
}

__global__ void final_k(const float* __restrict__ O, const float* __restrict__ bias,
                        float* __restrict__ outA, int Dtot, int Dpad, int outStride,
                        int doElu) {
    long t = (long)blockIdx.x * blockDim.x + threadIdx.x;
    if (t >= (long)NN * Dtot) return;
    int n = (int)(t / Dtot), d = (int)(t - (long)n * Dtot);
    float v = O[(long)n * Dpad + d] + bias[d];
    if (doElu) v = (v > 0.0f) ? v : (expf(v) - 1.0f);
    outA[(long)n * outStride + d] = v;
}

__global__ void cnt_k(const int* __restrict__ batch, float* __restrict__ cnt) {
    int n = blockIdx.x * blockDim.x + threadIdx.x;
    if (n < NN) atomicAdd(&cnt[batch[n]], 1.0f);
}

__global__ void pool_k(const int* __restrict__ batch, const float* __restrict__ O,
                       float* __restrict__ PL) {
    int n = blockIdx.x;
    int d = blockIdx.y * blockDim.x + threadIdx.x;
    if (d >= D3) return;
    atomicAdd(&PL[(long)batch[n] * D3P + d], O[(long)n * D3P + d]);
}

__global__ void head_k(const float* __restrict__ PL, const float* __restrict__ CNT,
                       const float* __restrict__ wp, const float* __restrict__ bp,
                       float* __restrict__ out) {
    int t = blockIdx.x * blockDim.x + threadIdx.x;
    if (t >= BB * NCLS) return;
    int b = t / NCLS, c = t - b * NCLS;
    float inv = 1.0f / fmaxf(CNT[b], 1.0f);
    float acc = bp[c];
    for (int j = 0; j < OUT3; ++j) {
        float gm = 0.0f;
        #pragma unroll
        for (int h = 0; h < H3; ++h) gm += PL[(long)b * D3P + h * OUT3 + j];
        acc += gm * inv * (1.0f / 6.0f) * wp[j * NCLS + c];
    }
    out[t] = acc;
}

// ---------------- host ----------------
static inline int cdiv(long a, long b) { return (int)((a + b - 1) / b); }

extern "C" void kernel_launch(void* const* d_in, const int* in_sizes, int n_in,
                              void* d_out, int out_size, void* d_ws, size_t ws_size,
                              hipStream_t stream) {
    const int*   x      = (const int*)d_in[0];
    const int*   depth  = (const int*)d_in[1];
    const int*   ei     = (const int*)d_in[2];   // [2,E]: row0=src, row1=dst
    const int*   batch  = (const int*)d_in[3];
    const float* nemb   = (const float*)d_in[4];
    const float* demb   = (const float*)d_in[5];
    const float* w1     = (const float*)d_in[6];
    const float* as1    = (const float*)d_in[7];
    const float* ad1    = (const float*)d_in[8];
    const float* b1     = (const float*)d_in[9];
    const float* w2     = (const float*)d_in[10];
    const float* as2    = (const float*)d_in[11];
    const float* ad2    = (const float*)d_in[12];
    const float* b2     = (const float*)d_in[13];
    const float* w3     = (const float*)d_in[14];
    const float* as3    = (const float*)d_in[15];
    const float* ad3    = (const float*)d_in[16];
    const float* b3     = (const float*)d_in[17];
    const float* wp     = (const float*)d_in[18];
    const float* bp     = (const float*)d_in[19];
    float*       out    = (float*)d_out;

    // workspace layout (floats)
    float* ws = (float*)d_ws;
    long off = 0;
    float* A   = ws + off; off += (long)NN * D1;     // layer inputs (layer1 packed N x 64)
    float* Hf  = ws + off; off += (long)NN * D3P;    // transformed features
    float* O   = ws + off; off += (long)NN * D3P;    // aggregated output
    float* Eb  = ws + off; off += (long)ETOT * HP;   // edge logits / exp
    float* Mx  = ws + off; off += (long)NN * HP;
    float* Sx  = ws + off; off += (long)NN * HP;
    float* ES  = ws + off; off += (long)NN * HP;
    float* ED  = ws + off; off += (long)NN * HP;
    float* W3P = ws + off; off += 256L * D3P;
    float* PL  = ws + off; off += (long)BB * D3P;
    float* CNT = ws + off; off += BB;

    const int T = 256;

    // pad conv3 weights to 736 columns
    padw3_k<<<cdiv(256L * D3P, T), T, 0, stream>>>(w3, W3P);
    // node encoder
    encode_k<<<cdiv((long)NN * EMB, T), T, 0, stream>>>(x, depth, nemb, demb, A);

    struct Layer {
        const float *W, *as, *ad, *b;
        int K, Hh, OD, Dpad, elu;
    };
    Layer layers[3] = {
        { w1,  as1, ad1, b1, EMB, 4, EMB,  D1,  1 },
        { w2,  as2, ad2, b2, D1,  4, EMB,  D1,  1 },
        { W3P, as3, ad3, b3, D1,  H3, OUT3, D3P, 0 },
    };

    for (int L = 0; L < 3; ++L) {
        const Layer& P = layers[L];
        int Dtot = P.Hh * P.OD;

        // 1) GEMM Hf = A @ W  (fp32 WMMA, NT tiles per wave)
        {
            long waves = (NN / 16) * (long)((P.Dpad / 16) / NT);
            int blocks = cdiv(waves, 8);   // 8 waves per 256-thread block
            gemm_wmma_f32<<<blocks, 256, 0, stream>>>(A, P.W, Hf, P.K, P.Dpad);
        }
        // 2) per-node attention logits
        logits_k<<<cdiv((long)NN * P.Hh, T), T, 0, stream>>>(Hf, P.as, P.ad, ES, ED,
                                                             P.Hh, P.OD, P.Dpad);
        // 3) init reductions
        fill_k<<<cdiv((long)NN * HP, T), T, 0, stream>>>(Mx, -1e30f, (long)NN * HP);
        fill_k<<<cdiv((long)NN * HP, T), T, 0, stream>>>(Sx, 0.0f, (long)NN * HP);
        fill_k<<<cdiv((long)NN * P.Dpad, T), T, 0, stream>>>(O, 0.0f, (long)NN * P.Dpad);
        // 4) edge logits + segment max
        edge_max_k<<<cdiv((long)ETOT * P.Hh, T), T, 0, stream>>>(ei, ES, ED, Eb, Mx, P.Hh);
        // 5) exp + segment sum
        edge_exp_k<<<cdiv((long)ETOT * P.Hh, T), T, 0, stream>>>(ei, Mx, Eb, Sx, P.Hh);
        // 6) alpha-weighted scatter add
        {
            dim3 g(ETOT, (unsigned)cdiv(Dtot, T));
            scatter_k<<<g, T, 0, stream>>>(ei, Hf, Eb, Sx, O, P.OD, P.Hh, P.Dpad);
        }
        // 7) bias (+ELU) finalize
        if (P.elu) {
            final_k<<<cdiv((long)NN * Dtot, T), T, 0, stream>>>(O, P.b, A, Dtot, P.Dpad, D1, 1);
        } else {
            final_k<<<cdiv((long)NN * Dtot, T), T, 0, stream>>>(O, P.b, O, Dtot, P.Dpad, P.Dpad, 0);
        }
    }

    // pooling
    fill_k<<<cdiv((long)BB * D3P, T), T, 0, stream>>>(PL, 0.0f, (long)BB * D3P);
    fill_k<<<cdiv((long)BB, T), T, 0, stream>>>(CNT, 0.0f, (long)BB);
    cnt_k<<<cdiv((long)NN, T), T, 0, stream>>>(batch, CNT);
    {
        dim3 g(NN, (unsigned)cdiv(D3, T));
        pool_k<<<g, T, 0, stream>>>(batch, O, PL);
    }
    // head
    head_k<<<cdiv((long)BB * NCLS, T), T, 0, stream>>>(PL, CNT, wp, bp, out);
}